// TorchFusedReshapedEmbeddingGatherEinsum_78383153152660
// MI455X (gfx1250) — compile-verified
//
#include <hip/hip_runtime.h>

typedef __attribute__((ext_vector_type(2))) float v2f;
typedef __attribute__((ext_vector_type(4))) float v4f;
typedef __attribute__((ext_vector_type(8))) float v8f;

#define B_  4
#define T_  4096
#define I_  512
#define E_  16
#define K_  1024
#define J_  512

#define TILE_M   256   // gathered rows per block (32 per wave)
#define TILE_N   64    // J columns per block
#define TILE_I   32    // inner (I) chunk staged in LDS
#define A_STRIDE 36    // padded LDS row stride (floats), conflict-free & 16B aligned
#define BP_STRIDE 132  // pair-interleaved B: floats per K-pair row (64 cols * 2 + pad)

__launch_bounds__(256, 1)
__global__ void gather_gemm_wmma_f32(const float* __restrict__ X,
                                     const long long* __restrict__ ind,
                                     const float* __restrict__ W,
                                     float* __restrict__ Y) {
  __shared__ float sA[TILE_M * A_STRIDE];         // 256 x 36 floats = 36 KB
  __shared__ float sB[(TILE_I / 2) * BP_STRIDE];  // 16 x 132 floats = 8.25 KB

  const int t    = threadIdx.x;
  const int wave = t >> 5;            // 0..7  (wave32)
  const int lane = t & 31;
  const int l15  = lane & 15;
  const int hi   = lane >> 4;         // 0 for lanes 0-15, 1 for lanes 16-31
  const int klo  = hi << 1;           // K offset of this lane half in an f32 A/B frag

  const int k0 = blockIdx.x * TILE_M; // gathered-row tile origin (M)
  const int j0 = blockIdx.y * TILE_N; // J tile origin (N)
  const int be = blockIdx.z;          // b*E + e
  const int bi = be >> 4;             // / E_
  const int e  = be & 15;

  // ---- fixed per-thread cooperative-load geometry ----
  // A tile: 256 rows x 32 floats = 2048 float4; thread handles f = t + 256*q, q=0..7
  const int rA = t >> 3;              // base row 0..31; rows rA + 32q
  const int cA = (t & 7) << 2;        // float4 column offset within row
  const long long* indBase = ind + (size_t)be * K_ + k0;
  int offA[8];                        // 32-bit element offsets into X (fits: 8.4M elems)
  int ldsA[8];
#pragma unroll
  for (int q = 0; q < 8; ++q) {
    const int r   = rA + q * 32;
    const int row = (int)indBase[r];
    offA[q] = (bi * T_ + row) * I_ + cA;
    ldsA[q] = r * A_STRIDE + cA;
  }
  // B tile: 32 rows x 64 floats = 512 float4; thread handles f = t + 256*q, q=0..1
  const int cB = (t & 15) << 2;       // float column in row
  int offB[2];                        // element offsets into W (fits: 4.2M elems)
  int irB[2];
#pragma unroll
  for (int q = 0; q < 2; ++q) {
    const int ir = (t >> 4) + q * 16; // I-row within chunk, 0..31
    irB[q]  = ir;
    offB[q] = (e * I_ + ir) * J_ + j0 + cB;
  }

  v8f aL0 = {}, aL1 = {}, aL2 = {}, aL3 = {};   // rows m0..m0+15
  v8f aH0 = {}, aH1 = {}, aH2 = {}, aH3 = {};   // rows m0+16..m0+31
  const int m0 = wave * 32;

  for (int chunk = 0; chunk < I_ / TILE_I; ++chunk) {
    const int i0 = chunk * TILE_I;

    // stage global loads into registers (overlaps previous chunk's compute)
    v4f ra[8];
#pragma unroll
    for (int q = 0; q < 8; ++q) ra[q] = *(const v4f*)(X + offA[q] + i0);
    const v4f rb0 = *(const v4f*)(W + offB[0] + (size_t)i0 * J_);
    const v4f rb1 = *(const v4f*)(W + offB[1] + (size_t)i0 * J_);

    if (chunk + 1 < I_ / TILE_I) {    // emits global_prefetch_b8
      __builtin_prefetch(X + offA[0] + i0 + TILE_I, 0, 0);
      __builtin_prefetch(W + offB[0] + (size_t)(i0 + TILE_I) * J_, 0, 0);
    }

    __syncthreads();   // previous chunk's LDS reads complete
#pragma unroll
    for (int q = 0; q < 8; ++q) *(v4f*)(sA + ldsA[q]) = ra[q];
    // pair-interleaved B: (i', j) -> (i'>>1)*BP_STRIDE + j*2 + (i'&1)
    {
      float* d0 = sB + (irB[0] >> 1) * BP_STRIDE + (cB << 1) + (irB[0] & 1);
      float* d1 = sB + (irB[1] >> 1) * BP_STRIDE + (cB << 1) + (irB[1] & 1);
      d0[0] = rb0[0]; d0[2] = rb0[1]; d0[4] = rb0[2]; d0[6] = rb0[3];
      d1[0] = rb1[0]; d1[2] = rb1[1]; d1[4] = rb1[2]; d1[6] = rb1[3];
    }
    __syncthreads();   // tiles visible to all waves

#pragma unroll
    for (int ii = 0; ii < TILE_I; ii += 4) {
      // A fragments 16x4 f32: lanes 0-15 hold K=ii+0/ii+1, lanes 16-31 K=ii+2/ii+3
      const v2f aLo = *(const v2f*)(sA + (m0 + l15) * A_STRIDE + ii + klo);
      const v2f aHi = *(const v2f*)(sA + (m0 + 16 + l15) * A_STRIDE + ii + klo);
      // B pair row: lanes 0-15 -> pair ii/2, lanes 16-31 -> pair ii/2+1
      const float* bRow = sB + ((ii >> 1) + hi) * BP_STRIDE + (l15 << 1);
      {
        const v2f b = *(const v2f*)(bRow + 0 * 32);   // N sub-tile 0
        aL0 = __builtin_amdgcn_wmma_f32_16x16x4_f32(false, aLo, false, b, (short)0, aL0, false, false);
        aH0 = __builtin_amdgcn_wmma_f32_16x16x4_f32(false, aHi, false, b, (short)0, aH0, false, false);
      }
      {
        const v2f b = *(const v2f*)(bRow + 1 * 32);   // N sub-tile 1
        aL1 = __builtin_amdgcn_wmma_f32_16x16x4_f32(false, aLo, false, b, (short)0, aL1, false, false);
        aH1 = __builtin_amdgcn_wmma_f32_16x16x4_f32(false, aHi, false, b, (short)0, aH1, false, false);
      }
      {
        const v2f b = *(const v2f*)(bRow + 2 * 32);   // N sub-tile 2
        aL2 = __builtin_amdgcn_wmma_f32_16x16x4_f32(false, aLo, false, b, (short)0, aL2, false, false);
        aH2 = __builtin_amdgcn_wmma_f32_16x16x4_f32(false, aHi, false, b, (short)0, aH2, false, false);
      }
      {
        const v2f b = *(const v2f*)(bRow + 3 * 32);   // N sub-tile 3
        aL3 = __builtin_amdgcn_wmma_f32_16x16x4_f32(false, aLo, false, b, (short)0, aL3, false, false);
        aH3 = __builtin_amdgcn_wmma_f32_16x16x4_f32(false, aHi, false, b, (short)0, aH3, false, false);
      }
    }
  }

  // ---- store D tiles: VGPR r -> (M = r + 8*hi, N = l15) ----
  const int mOff = hi << 3;
  float* outL = Y + ((size_t)be * K_ + k0 + m0 + mOff) * J_ + j0 + l15;
  float* outH = outL + (size_t)16 * J_;
#pragma unroll
  for (int r = 0; r < 8; ++r) {
    float* oL = outL + (size_t)r * J_;
    float* oH = outH + (size_t)r * J_;
    oL[0]  = aL0[r];  oL[16] = aL1[r];  oL[32] = aL2[r];  oL[48] = aL3[r];
    oH[0]  = aH0[r];  oH[16] = aH1[r];  oH[32] = aH2[r];  oH[48] = aH3[r];
  }
}

extern "C" void kernel_launch(void* const* d_in, const int* in_sizes, int n_in,
                              void* d_out, int out_size, void* d_ws, size_t ws_size,
                              hipStream_t stream) {
  (void)in_sizes; (void)n_in; (void)out_size; (void)d_ws; (void)ws_size;
  const float*     X   = (const float*)d_in[0];
  const long long* ind = (const long long*)d_in[1];   // int64 indices
  const float*     W   = (const float*)d_in[2];
  float*           Y   = (float*)d_out;

  dim3 grid(K_ / TILE_M, J_ / TILE_N, B_ * E_);   // 4 x 8 x 64
  dim3 block(256, 1, 1);                          // 8 waves (wave32)
  gather_gemm_wmma_f32<<<grid, block, 0, stream>>>(X, ind, W, Y);
}